// SAGE_30365418783421
// MI455X (gfx1250) — compile-verified
//
#include <hip/hip_runtime.h>
#include <hip/hip_bf16.h>
#include <math.h>

// ---------------------------------------------------------------------------
// 2-layer hetero GraphSAGE forward for MI455X (gfx1250, wave32).
//  - scatter-mean via f32 global atomics (L2-resident: tables fit in 192MB L2)
//  - fused GEMM+bias+L2norm+leakyReLU using native fp32 WMMA (16x16x4 f32)
// ---------------------------------------------------------------------------

typedef float v2f __attribute__((ext_vector_type(2)));
typedef float v8f __attribute__((ext_vector_type(8)));

#define DIN 128
#define HID 256
#define TM 16          // rows (M) per block
#define NTHREADS 256   // 8 wave32 per block -> 8 x 32 = 256 output columns

// --------------------------- workspace zero fill ---------------------------
__global__ void sage_zero(float4* __restrict__ p, long long n4) {
  long long i = (long long)blockIdx.x * blockDim.x + threadIdx.x;
  if (i < n4) p[i] = make_float4(0.f, 0.f, 0.f, 0.f);
}

// ------------------ scatter: agg[dst] += x[src]; deg[dst]++ ----------------
template <int D>
__global__ void sage_scatter(const float* __restrict__ xsrc,
                             const int* __restrict__ ei,   // [2,E]: row0=src, row1=dst
                             float* __restrict__ agg,
                             float* __restrict__ deg,
                             int E) {
  constexpr int CH = D / 4;                     // float4 chunks per row
  int t = blockIdx.x * blockDim.x + threadIdx.x;
  int e = t / CH;
  int c = t % CH;
  if (e >= E) return;
  int src = ei[e];
  int dst = ei[E + e];
  const float4 v = *reinterpret_cast<const float4*>(xsrc + (size_t)src * D + c * 4);
  float* p = agg + (size_t)dst * D + c * 4;
  atomicAdd(p + 0, v.x);
  atomicAdd(p + 1, v.y);
  atomicAdd(p + 2, v.z);
  atomicAdd(p + 3, v.w);
  if (c == 0) atomicAdd(deg + dst, 1.0f);
}

// ------- fused: out = leaky(normalize((agg/deg)@Wl + b + xdst@Wr)) ---------
template <int D>
__global__ __launch_bounds__(NTHREADS)
void sage_gemm_norm(const float* __restrict__ agg,   // [n, D]
                    const float* __restrict__ deg,   // [n]
                    const float* __restrict__ xdst,  // [n, D] (may alias out!)
                    const float* __restrict__ Wl,    // [D, HID]
                    const float* __restrict__ Wr,    // [D, HID]
                    const float* __restrict__ bias,  // [HID]
                    float* __restrict__ out,         // [n, HID]
                    int n) {
  constexpr int K2 = 2 * D;       // concatenated K: [mean | x_dst]
  constexpr int LS = K2 + 4;      // padded LDS stride (stride%64==4 -> no bank conflicts)
  __shared__ float As[TM * LS];
  __shared__ float rowsum[TM];
  __shared__ float invdeg[TM];

  const int tid  = threadIdx.x;
  const int row0 = blockIdx.x * TM;

  if (tid < TM) {
    int g = row0 + tid;
    invdeg[tid] = (g < n) ? 1.0f / fmaxf(deg[g], 1.0f) : 0.0f;
    rowsum[tid] = 0.0f;
  }
  __syncthreads();

  // Stage A' = [agg/deg | x_dst]  (TM x 2D) into LDS; reads complete here,
  // which makes out==xdst aliasing safe (block only writes its own rows).
  for (int idx = tid; idx < TM * K2; idx += NTHREADS) {
    int r = idx / K2, j = idx - r * K2;
    int g = row0 + r;
    float v = 0.0f;
    if (g < n)
      v = (j < D) ? agg[(size_t)g * D + j] * invdeg[r]
                  : xdst[(size_t)g * D + (j - D)];
    As[r * LS + j] = v;
  }
  __syncthreads();

  const int wave = tid >> 5;
  const int lane = tid & 31;
  const int hrow = lane & 15;               // A-row / B-col within a 16-tile
  const int koff = (lane < 16) ? 0 : 2;     // ISA 16x4 layout: lanes16-31 hold K+2,K+3
  const int n0   = wave * 32;               // this wave's two 16-col tiles
  const int n1   = n0 + 16;

  v8f acc0 = {};
  v8f acc1 = {};

  const float* lds_a = &As[hrow * LS];
  for (int half = 0; half < 2; ++half) {
    const float* __restrict__ W = half ? Wr : Wl;
    const int kb = half * D;
    for (int k = 0; k < D; k += 4) {
      v2f a, b0, b1;
      a.x = lds_a[kb + k + koff];
      a.y = lds_a[kb + k + koff + 1];
      const float* wp = W + (size_t)(k + koff) * HID;
      b0.x = wp[n0 + hrow];
      b0.y = wp[HID + n0 + hrow];
      b1.x = wp[n1 + hrow];
      b1.y = wp[HID + n1 + hrow];
      acc0 = __builtin_amdgcn_wmma_f32_16x16x4_f32(false, a, false, b0,
                                                   (short)0, acc0, false, false);
      acc1 = __builtin_amdgcn_wmma_f32_16x16x4_f32(false, a, false, b1,
                                                   (short)0, acc1, false, false);
    }
  }

  // Epilogue: bias, per-row squared-norm partials (lane holds rows v(+8), 2 cols)
  const float bv0 = bias[n0 + hrow];
  const float bv1 = bias[n1 + hrow];
  float s[8];
#pragma unroll
  for (int v = 0; v < 8; ++v) {
    acc0[v] += bv0;
    acc1[v] += bv1;
    s[v] = acc0[v] * acc0[v] + acc1[v] * acc1[v];
  }
  // reduce the 16 lanes that share each row (xor<=8 stays inside lane halves)
#pragma unroll
  for (int off = 1; off < 16; off <<= 1) {
#pragma unroll
    for (int v = 0; v < 8; ++v) s[v] += __shfl_xor(s[v], off, 32);
  }
  if (hrow == 0) {  // lane 0 -> rows 0..7, lane 16 -> rows 8..15
    int rbase = (lane < 16) ? 0 : 8;
#pragma unroll
    for (int v = 0; v < 8; ++v) atomicAdd(&rowsum[rbase + v], s[v]);
  }
  __syncthreads();

  const int rb = (lane < 16) ? 0 : 8;
#pragma unroll
  for (int v = 0; v < 8; ++v) {
    int m = rb + v;
    int g = row0 + m;
    if (g >= n) continue;
    float sc = 1.0f / fmaxf(sqrtf(rowsum[m]), 1e-12f);
    float y0 = acc0[v] * sc;
    float y1 = acc1[v] * sc;
    y0 = (y0 > 0.0f) ? y0 : 0.01f * y0;   // leaky_relu(0.01)
    y1 = (y1 > 0.0f) ? y1 : 0.01f * y1;
    out[(size_t)g * HID + n0 + hrow] = y0;
    out[(size_t)g * HID + n1 + hrow] = y1;
  }
}

// ---------------------------------------------------------------------------
extern "C" void kernel_launch(void* const* d_in, const int* in_sizes, int n_in,
                              void* d_out, int out_size, void* d_ws, size_t ws_size,
                              hipStream_t stream) {
  (void)n_in; (void)out_size; (void)ws_size;

  const float* xa    = (const float*)d_in[0];
  const float* xb    = (const float*)d_in[1];
  const int*   ei_ab = (const int*)d_in[2];
  const int*   ei_ba = (const int*)d_in[3];
  const float* Wl0_ab = (const float*)d_in[4];
  const float* Wr0_ab = (const float*)d_in[5];
  const float* b0_ab  = (const float*)d_in[6];
  const float* Wl0_ba = (const float*)d_in[7];
  const float* Wr0_ba = (const float*)d_in[8];
  const float* b0_ba  = (const float*)d_in[9];
  const float* Wl1_ab = (const float*)d_in[10];
  const float* Wr1_ab = (const float*)d_in[11];
  const float* b1_ab  = (const float*)d_in[12];
  const float* Wl1_ba = (const float*)d_in[13];
  const float* Wr1_ba = (const float*)d_in[14];
  const float* b1_ba  = (const float*)d_in[15];

  const int NA = in_sizes[0] / DIN;
  const int NB = in_sizes[1] / DIN;
  const int E  = in_sizes[2] / 2;

  // workspace: agg_a [NA,HID] | agg_b [NB,HID] | deg_a [NA] | deg_b [NB]
  float* ws    = (float*)d_ws;
  float* agg_a = ws;
  float* agg_b = agg_a + (size_t)NA * HID;
  float* deg_a = agg_b + (size_t)NB * HID;
  float* deg_b = deg_a + NA;
  const long long ws_floats = (long long)(NA + NB) * HID + NA + NB;
  const long long ws_f4     = (ws_floats + 3) / 4;

  float* ya = (float*)d_out;               // final xa [NA,HID]
  float* yb = ya + (size_t)NA * HID;       // final xb [NB,HID]

  const dim3 blk(NTHREADS);
  const int zgrid  = (int)((ws_f4 + NTHREADS - 1) / NTHREADS);
  const int sgrid0 = (E * (DIN / 4) + NTHREADS - 1) / NTHREADS;
  const int sgrid1 = (E * (HID / 4) + NTHREADS - 1) / NTHREADS;
  const int ggrid_a = (NA + TM - 1) / TM;
  const int ggrid_b = (NB + TM - 1) / TM;

  // ---------------- layer 0 (din = 128) ----------------
  sage_zero<<<zgrid, blk, 0, stream>>>((float4*)d_ws, ws_f4);
  sage_scatter<DIN><<<sgrid0, blk, 0, stream>>>(xa, ei_ab, agg_b, deg_b, E);
  sage_scatter<DIN><<<sgrid0, blk, 0, stream>>>(xb, ei_ba, agg_a, deg_a, E);
  sage_gemm_norm<DIN><<<ggrid_b, blk, 0, stream>>>(agg_b, deg_b, xb,
                                                   Wl0_ab, Wr0_ab, b0_ab, yb, NB);
  sage_gemm_norm<DIN><<<ggrid_a, blk, 0, stream>>>(agg_a, deg_a, xa,
                                                   Wl0_ba, Wr0_ba, b0_ba, ya, NA);

  // ---------------- layer 1 (din = 256) ----------------
  // Both scatters read layer-0 outputs BEFORE either GEMM overwrites them.
  sage_zero<<<zgrid, blk, 0, stream>>>((float4*)d_ws, ws_f4);
  sage_scatter<HID><<<sgrid1, blk, 0, stream>>>(ya, ei_ab, agg_b, deg_b, E);
  sage_scatter<HID><<<sgrid1, blk, 0, stream>>>(yb, ei_ba, agg_a, deg_a, E);
  // out aliases x_dst: safe, each block stages its own 16 rows to LDS first.
  sage_gemm_norm<HID><<<ggrid_b, blk, 0, stream>>>(agg_b, deg_b, yb,
                                                   Wl1_ab, Wr1_ab, b1_ab, yb, NB);
  sage_gemm_norm<HID><<<ggrid_a, blk, 0, stream>>>(agg_a, deg_a, ya,
                                                   Wl1_ba, Wr1_ba, b1_ba, ya, NA);
}